// LunarCausalAttention_38637525794826
// MI455X (gfx1250) — compile-verified
//
#include <hip/hip_runtime.h>
#include <math.h>
#include <stdint.h>

// ---------------------------------------------------------------------------
// LunarCausalAttention on MI455X (gfx1250, wave32, WMMA).
// fp32 WMMA (V_WMMA_F32_16X16X4_F32) everywhere: the problem moves ~50MB
// (fits in 192MB L2) against 23.3 TB/s HBM, so it is bandwidth-bound and fp32
// matrix throughput is far above the roofline need -> keep full precision.
// The causal exclusive-cumsum stages are chunked prefix scans whose chunk
// totals and in-chunk causal parts are 16x16 WMMA tiles (linear time).
// The two 4096x1024x1024 projection GEMMs + output GEMM use block-level LDS
// tiling fed by CDNA5 async global->LDS copies (ASYNCcnt / s_wait_asynccnt).
// ---------------------------------------------------------------------------

#define N_SEQ  2048
#define BATCH  2
#define EMB    1024
#define NH     16
#define PQN    16
#define HD     64
#define BH     (BATCH * NH)     // 32
#define NC     (N_SEQ / 16)     // 128 chunks of 16 rows
#define SCALING 0.125f          // hd^-0.5

#define LDSTRIDE 36             // padded float stride for 32-wide k-tiles

typedef __attribute__((ext_vector_type(2))) float v2f;
typedef __attribute__((ext_vector_type(8))) float v8f;

__device__ __forceinline__ v8f wmma_f32(v2f a, v2f b, v8f c) {
  // D = A(16x4 f32) * B(4x16 f32) + C(16x16 f32)
  return __builtin_amdgcn_wmma_f32_16x16x4_f32(
      /*neg_a=*/false, a, /*neg_b=*/false, b,
      /*c_mod=*/(short)0, c, /*reuse_a=*/false, /*reuse_b=*/false);
}

__device__ __forceinline__ v8f v8zero() {
  v8f z = {0.f, 0.f, 0.f, 0.f, 0.f, 0.f, 0.f, 0.f};
  return z;
}

// CDNA5 async global->LDS copy (tracked by ASYNCcnt), 16 bytes per lane.
__device__ __forceinline__ void async_g2l_b128(uint32_t lds_byte_addr,
                                               const float* gptr) {
  asm volatile("global_load_async_to_lds_b128 %0, %1, off"
               :
               : "v"(lds_byte_addr),
                 "v"((unsigned long long)(uintptr_t)gptr)
               : "memory");
}
__device__ __forceinline__ void wait_async0() {
  asm volatile("s_wait_asynccnt 0x0" ::: "memory");
}

// ---------------------------------------------------------------------------
// 1) pq projection (tiny 32x1024x1024): scalar dot products.
// ---------------------------------------------------------------------------
__global__ void pq_proj_kernel(const float* __restrict__ pquery,
                               const float* __restrict__ Wpq,
                               const float* __restrict__ bpq,
                               float* __restrict__ pq) {
  int idx = blockIdx.x * blockDim.x + threadIdx.x;  // BH*PQN*HD = 32768
  int d  = idx & (HD - 1);
  int p  = (idx >> 6) & (PQN - 1);
  int bh = idx >> 10;
  int bb = bh / NH;
  int h  = bh % NH;
  int row = h * HD + d;
  const float* x = pquery + (p * BATCH + bb) * EMB;
  const float* w = Wpq + row * EMB;
  float acc = bpq[row];
  for (int e = 0; e < EMB; ++e) acc += x[e] * w[e];
  pq[idx] = acc * SCALING;
}

// ---------------------------------------------------------------------------
// 2) K/V projection (WMMA + async LDS staging):
//    dst[(b*NH+h)][i][d] = query(4096x1024) @ W^T + bias
//    256 threads = 8 waves per block; block tile 64 rows x 64 cols;
//    K staged 32-deep through LDS via global_load_async_to_lds_b128.
// ---------------------------------------------------------------------------
__global__ void kv_proj_kernel(const float* __restrict__ X,
                               const float* __restrict__ W,
                               const float* __restrict__ bias,
                               float* __restrict__ dst) {
  __shared__ float lA[64 * LDSTRIDE];
  __shared__ float lB[64 * LDSTRIDE];

  int lane = threadIdx.x & 31;
  int half = lane >> 4, r = lane & 15;
  int wib  = threadIdx.x >> 5;  // 0..7
  int wr   = wib >> 1;          // 0..3 (16-row tile)
  int wc   = wib & 1;           // 0..1 (32-col slab = two 16-col tiles)

  int ctb = blockIdx.x & 15;          // 1024/64 = 16 col blocks
  int rtb = blockIdx.x >> 4;          // 4096/64 = 64 row blocks
  int row0 = rtb * 64, col0 = ctb * 64;

  uint32_t baseA = (uint32_t)(uintptr_t)&lA[0];
  uint32_t baseB = (uint32_t)(uintptr_t)&lB[0];

  int rowL = threadIdx.x >> 2;  // 0..63 (staging row)
  int seg  = threadIdx.x & 3;   // 0..3  (8-float segment)
  uint32_t la = baseA + (uint32_t)(rowL * LDSTRIDE + seg * 8) * 4u;
  uint32_t lb = baseB + (uint32_t)(rowL * LDSTRIDE + seg * 8) * 4u;
  const float* gaBase = X + (size_t)(row0 + rowL) * EMB + seg * 8;
  const float* gbBase = W + (size_t)(col0 + rowL) * EMB + seg * 8;

  v8f acc0 = v8zero(), acc1 = v8zero();
  for (int k0 = 0; k0 < EMB; k0 += 32) {
    __syncthreads();  // previous tile fully consumed
    async_g2l_b128(la,      gaBase + k0);
    async_g2l_b128(la + 16, gaBase + k0 + 4);
    async_g2l_b128(lb,      gbBase + k0);
    async_g2l_b128(lb + 16, gbBase + k0 + 4);
    wait_async0();
    __syncthreads();  // staged tile visible to all waves
#pragma unroll
    for (int kk = 0; kk < 32; kk += 4) {
      int dd = kk + 2 * half;
      v2f a  = *(const v2f*)&lA[(wr * 16 + r) * LDSTRIDE + dd];
      v2f b0 = *(const v2f*)&lB[(wc * 32 + r) * LDSTRIDE + dd];
      v2f b1 = *(const v2f*)&lB[(wc * 32 + 16 + r) * LDSTRIDE + dd];
      acc0 = wmma_f32(a, b0, acc0);
      acc1 = wmma_f32(a, b1, acc1);
    }
  }

#pragma unroll
  for (int ti = 0; ti < 2; ++ti) {
    int colg = col0 + wc * 32 + ti * 16 + r;
    float bval = bias[colg];
    int h = colg >> 6, d = colg & 63;
    v8f acc = ti ? acc1 : acc0;
#pragma unroll
    for (int t = 0; t < 8; ++t) {
      int m = row0 + wr * 16 + t + 8 * half;
      int i = m >> 1, bb = m & 1;
      dst[(((size_t)(bb * NH + h)) * N_SEQ + i) * HD + d] = acc[t] + bval;
    }
  }
}

// ---------------------------------------------------------------------------
// 3) pattn (WMMA): pattn[bh][i][p] = qh[i,b,h,:] . pq[bh][p][:]   (K = 64)
// ---------------------------------------------------------------------------
__global__ void pattn_kernel(const float* __restrict__ query,
                             const float* __restrict__ pq,
                             float* __restrict__ pattn) {
  int wid  = (blockIdx.x * blockDim.x + threadIdx.x) >> 5;
  int lane = threadIdx.x & 31;
  int half = lane >> 4, r = lane & 15;
  int rt = wid % NC;
  int bh = wid / NC;
  int bb = bh / NH, h = bh % NH;
  int base = rt * 16;

  const float* aptr = query + ((size_t)(base + r) * BATCH + bb) * EMB + h * HD;
  const float* bptr = pq + (size_t)(bh * PQN + r) * HD;
  v8f acc = v8zero();
  for (int kk = 0; kk < HD; kk += 4) {
    v2f a = *(const v2f*)(aptr + kk + 2 * half);
    v2f b = *(const v2f*)(bptr + kk + 2 * half);
    acc = wmma_f32(a, b, acc);
  }
#pragma unroll
  for (int t = 0; t < 8; ++t) {
    int i = base + t + 8 * half;
    pattn[((size_t)bh * N_SEQ + i) * PQN + r] = acc[t];
  }
}

// ---------------------------------------------------------------------------
// 4) per-(bh,p) column: global max, Eexp = exp(pattn-m) (in place),
//    S = exclusive cumsum over i. One 256-thread block per column.
// ---------------------------------------------------------------------------
__global__ void softmax_scan_kernel(float* __restrict__ pattn,
                                    float* __restrict__ sexc) {
  int bh = blockIdx.x >> 4;
  int p  = blockIdx.x & 15;
  int t  = threadIdx.x;
  float* col = pattn + (size_t)bh * N_SEQ * PQN + p;

  const int PER = N_SEQ / 256;  // 8
  float v[PER];
  float lmax = -3.0e38f;
  int i0 = t * PER;
#pragma unroll
  for (int u = 0; u < PER; ++u) {
    v[u] = col[(size_t)(i0 + u) * PQN];
    lmax = fmaxf(lmax, v[u]);
  }
  __shared__ float red[256];
  red[t] = lmax;
  __syncthreads();
  for (int s = 128; s > 0; s >>= 1) {
    if (t < s) red[t] = fmaxf(red[t], red[t + s]);
    __syncthreads();
  }
  float m = red[0];
  __syncthreads();

  float e[PER];
  float tot = 0.f;
#pragma unroll
  for (int u = 0; u < PER; ++u) {
    e[u] = __expf(v[u] - m);
    tot += e[u];
  }
  red[t] = tot;
  __syncthreads();
  for (int off = 1; off < 256; off <<= 1) {
    float x = (t >= off) ? red[t - off] : 0.f;
    __syncthreads();
    red[t] += x;
    __syncthreads();
  }
  float pre = red[t] - tot;  // exclusive prefix of this thread's segment
#pragma unroll
  for (int u = 0; u < PER; ++u) {
    sexc[((size_t)bh * N_SEQ + i0 + u) * PQN + p] = pre;
    col[(size_t)(i0 + u) * PQN] = e[u];
    pre += e[u];
  }
}

// ---------------------------------------------------------------------------
// 5) chunk totals stage 1 (WMMA): G_c[d][p] = sum_{j in c} k[j,d]*E[j,p]
// ---------------------------------------------------------------------------
__global__ void chunk_g_kernel(const float* __restrict__ kbuf,
                               const float* __restrict__ eexp,
                               float* __restrict__ gbuf) {
  int wid  = (blockIdx.x * blockDim.x + threadIdx.x) >> 5;
  int lane = threadIdx.x & 31;
  int half = lane >> 4, r = lane & 15;
  int c  = wid % NC;
  int bh = wid / NC;
  int base = c * 16;

  v8f acc[4] = {v8zero(), v8zero(), v8zero(), v8zero()};
  for (int kk = 0; kk < 16; kk += 4) {
    int j0 = base + kk + 2 * half;
    v2f b = {eexp[((size_t)bh * N_SEQ + j0) * PQN + r],
             eexp[((size_t)bh * N_SEQ + j0 + 1) * PQN + r]};
    const float* kr0 = kbuf + ((size_t)bh * N_SEQ + j0) * HD;
    const float* kr1 = kr0 + HD;
#pragma unroll
    for (int tI = 0; tI < 4; ++tI) {
      v2f a = {kr0[16 * tI + r], kr1[16 * tI + r]};
      acc[tI] = wmma_f32(a, b, acc[tI]);
    }
  }
#pragma unroll
  for (int tI = 0; tI < 4; ++tI)
#pragma unroll
    for (int vt = 0; vt < 8; ++vt) {
      int d = 16 * tI + vt + 8 * half;
      gbuf[(((size_t)(bh * NC + c)) * HD + d) * PQN + r] = acc[tI][vt];
    }
}

// ---------------------------------------------------------------------------
// 6) exclusive prefix over the 128 chunks, in place; one lane per (bh, elem).
// ---------------------------------------------------------------------------
__global__ void chunk_scan_kernel(float* __restrict__ buf) {
  int idx = blockIdx.x * blockDim.x + threadIdx.x;  // BH*1024
  int bh = idx >> 10;
  int e  = idx & 1023;
  float* ptr = buf + (size_t)bh * NC * 1024 + e;
  float carry = 0.f;
  for (int c = 0; c < NC; ++c) {
    float t = ptr[(size_t)c * 1024];
    ptr[(size_t)c * 1024] = carry;
    carry += t;
  }
}

// ---------------------------------------------------------------------------
// 7) stage 1 (WMMA): attn_weights tile = Qs_c @ PrefG_c + tril<(Qs_c K_c^T) @ E_c,
//    divide by Ssafe, row-softmax over P=16, write aw. One wave per (bh, c).
// ---------------------------------------------------------------------------
__global__ void stage1_kernel(const float* __restrict__ query,
                              const float* __restrict__ kbuf,
                              const float* __restrict__ eexp,
                              const float* __restrict__ sexc,
                              const float* __restrict__ gpref,
                              float* __restrict__ awbuf) {
  int wib  = threadIdx.x >> 5;                       // wave in block (0..3)
  int wid  = blockIdx.x * (blockDim.x >> 5) + wib;
  int lane = threadIdx.x & 31;
  int half = lane >> 4, r = lane & 15;
  int c  = wid % NC;
  int bh = wid / NC;
  int bb = bh / NH, h = bh % NH;
  int base = c * 16;

  __shared__ float sT[4][16][17];
  float(*s)[17] = sT[wib];

  const float* aptr = query + ((size_t)(base + r) * BATCH + bb) * EMB + h * HD;
  const float* gptr = gpref + (size_t)(bh * NC + c) * (HD * PQN);
  const float* kp   = kbuf + ((size_t)bh * N_SEQ + base + r) * HD;

  v8f accW = v8zero();  // attn_weights accumulator
  v8f accS = v8zero();  // in-chunk Q.K^T scores
  for (int kk = 0; kk < HD; kk += 4) {
    int dd = kk + 2 * half;
    v2f a = *(const v2f*)(aptr + dd);
    a = a * SCALING;
    v2f bg = {gptr[dd * PQN + r], gptr[(dd + 1) * PQN + r]};
    v2f bk = *(const v2f*)(kp + dd);
    accW = wmma_f32(a, bg, accW);
    accS = wmma_f32(a, bk, accS);
  }

  // D-layout scores -> LDS; reload in A-layout with strict-causal mask (j < i)
#pragma unroll
  for (int t = 0; t < 8; ++t) s[t + 8 * half][r] = accS[t];
  __syncthreads();
  for (int kk = 0; kk < 16; kk += 4) {
    int j0 = kk + 2 * half;
    v2f a = {(j0 < r) ? s[r][j0] : 0.f, (j0 + 1 < r) ? s[r][j0 + 1] : 0.f};
    int jg = base + j0;
    v2f b = {eexp[((size_t)bh * N_SEQ + jg) * PQN + r],
             eexp[((size_t)bh * N_SEQ + jg + 1) * PQN + r]};
    accW = wmma_f32(a, b, accW);
  }
  __syncthreads();

  // divide by Ssafe, stash into LDS for row-wise softmax over P
#pragma unroll
  for (int t = 0; t < 8; ++t) {
    int i = base + t + 8 * half;
    float S = sexc[((size_t)bh * N_SEQ + i) * PQN + r];
    s[t + 8 * half][r] = accW[t] / (S > 0.f ? S : 1.f);
  }
  __syncthreads();
  if (lane < 16) {
    float m = -3.0e38f;
#pragma unroll
    for (int p = 0; p < PQN; ++p) m = fmaxf(m, s[lane][p]);
    float sum = 0.f, ex[PQN];
#pragma unroll
    for (int p = 0; p < PQN; ++p) {
      ex[p] = __expf(s[lane][p] - m);
      sum += ex[p];
    }
    float inv = 1.f / sum;
#pragma unroll
    for (int p = 0; p < PQN; ++p) s[lane][p] = ex[p] * inv;
  }
  __syncthreads();
#pragma unroll
  for (int t = 0; t < 8; ++t) {
    int i = base + t + 8 * half;
    awbuf[((size_t)bh * N_SEQ + i) * PQN + r] = s[t + 8 * half][r];
  }
}

// ---------------------------------------------------------------------------
// 8) chunk totals stage 2 (WMMA): M_c[p][d] = sum_{j in c} E[j,p]*v[j,d]
// ---------------------------------------------------------------------------
__global__ void chunk_m_kernel(const float* __restrict__ eexp,
                               const float* __restrict__ vbuf,
                               float* __restrict__ mbuf) {
  int wid  = (blockIdx.x * blockDim.x + threadIdx.x) >> 5;
  int lane = threadIdx.x & 31;
  int half = lane >> 4, r = lane & 15;
  int c  = wid % NC;
  int bh = wid / NC;
  int base = c * 16;

  v8f acc[4] = {v8zero(), v8zero(), v8zero(), v8zero()};
  for (int kk = 0; kk < 16; kk += 4) {
    int j0 = base + kk + 2 * half;
    v2f a = {eexp[((size_t)bh * N_SEQ + j0) * PQN + r],
             eexp[((size_t)bh * N_SEQ + j0 + 1) * PQN + r]};
    const float* v0 = vbuf + ((size_t)bh * N_SEQ + j0) * HD;
    const float* v1 = v0 + HD;
#pragma unroll
    for (int tI = 0; tI < 4; ++tI) {
      v2f b = {v0[16 * tI + r], v1[16 * tI + r]};
      acc[tI] = wmma_f32(a, b, acc[tI]);
    }
  }
#pragma unroll
  for (int tI = 0; tI < 4; ++tI)
#pragma unroll
    for (int vt = 0; vt < 8; ++vt) {
      int p = vt + 8 * half;
      mbuf[(((size_t)(bh * NC + c)) * PQN + p) * HD + 16 * tI + r] = acc[tI][vt];
    }
}

// ---------------------------------------------------------------------------
// 9) stage 2 (WMMA): out tile = AW'_c @ PrefM_c + tril<(AW'_c E_c^T) @ V_c
// ---------------------------------------------------------------------------
__global__ void stage2_kernel(const float* __restrict__ awbuf,
                              const float* __restrict__ sexc,
                              const float* __restrict__ eexp,
                              const float* __restrict__ vbuf,
                              const float* __restrict__ mpref,
                              float* __restrict__ outpre) {
  int wib  = threadIdx.x >> 5;
  int wid  = blockIdx.x * (blockDim.x >> 5) + wib;
  int lane = threadIdx.x & 31;
  int half = lane >> 4, r = lane & 15;
  int c  = wid % NC;
  int bh = wid / NC;
  int base = c * 16;

  __shared__ float sT[4][16][17];
  float(*s)[17] = sT[wib];

  // A fragments of AW' (K = P = 16), cached across all 4 d-tiles
  v2f aA[4];
#pragma unroll
  for (int kq = 0; kq < 4; ++kq) {
    int p0 = kq * 4 + 2 * half;
    size_t rowoff = ((size_t)bh * N_SEQ + base + r) * PQN;
    float S0 = sexc[rowoff + p0];
    float S1 = sexc[rowoff + p0 + 1];
    aA[kq] = v2f{awbuf[rowoff + p0] / (S0 > 0.f ? S0 : 1.f),
                 awbuf[rowoff + p0 + 1] / (S1 > 0.f ? S1 : 1.f)};
  }

  // scores2 = AW'_c (16xP) @ E_c^T (Px16)
  v8f accS = v8zero();
#pragma unroll
  for (int kq = 0; kq < 4; ++kq) {
    int p0 = kq * 4 + 2 * half;
    v2f b = *(const v2f*)(eexp + ((size_t)bh * N_SEQ + base + r) * PQN + p0);
    accS = wmma_f32(aA[kq], b, accS);
  }
#pragma unroll
  for (int t = 0; t < 8; ++t) s[t + 8 * half][r] = accS[t];
  __syncthreads();
  v2f aS[4];
#pragma unroll
  for (int kq = 0; kq < 4; ++kq) {
    int j0 = kq * 4 + 2 * half;
    aS[kq] = v2f{(j0 < r) ? s[r][j0] : 0.f, (j0 + 1 < r) ? s[r][j0 + 1] : 0.f};
  }
  __syncthreads();

  const float* mp = mpref + (size_t)(bh * NC + c) * (PQN * HD);
#pragma unroll
  for (int tI = 0; tI < 4; ++tI) {
    v8f acc = v8zero();
#pragma unroll
    for (int kq = 0; kq < 4; ++kq) {
      int p0 = kq * 4 + 2 * half;
      v2f bm = {mp[p0 * HD + 16 * tI + r], mp[(p0 + 1) * HD + 16 * tI + r]};
      acc = wmma_f32(aA[kq], bm, acc);
      int j0 = base + kq * 4 + 2 * half;
      v2f bv2 = {vbuf[((size_t)bh * N_SEQ + j0) * HD + 16 * tI + r],
                 vbuf[((size_t)bh * N_SEQ + j0 + 1) * HD + 16 * tI + r]};
      acc = wmma_f32(aS[kq], bv2, acc);
    }
#pragma unroll
    for (int vt = 0; vt < 8; ++vt) {
      int i = base + vt + 8 * half;
      outpre[((size_t)bh * N_SEQ + i) * HD + 16 * tI + r] = acc[vt];
    }
  }
}

// ---------------------------------------------------------------------------
// 10) output projection (WMMA + async LDS staging):
//     attn = gather(outpre)(4096x1024) @ Wo^T + bo
// ---------------------------------------------------------------------------
__global__ void out_proj_kernel(const float* __restrict__ outpre,
                                const float* __restrict__ Wo,
                                const float* __restrict__ bo,
                                float* __restrict__ out) {
  __shared__ float lA[64 * LDSTRIDE];
  __shared__ float lB[64 * LDSTRIDE];

  int lane = threadIdx.x & 31;
  int half = lane >> 4, r = lane & 15;
  int wib  = threadIdx.x >> 5;
  int wr   = wib >> 1;
  int wc   = wib & 1;

  int ctb = blockIdx.x & 15;
  int rtb = blockIdx.x >> 4;
  int row0 = rtb * 64, col0 = ctb * 64;

  uint32_t baseA = (uint32_t)(uintptr_t)&lA[0];
  uint32_t baseB = (uint32_t)(uintptr_t)&lB[0];

  int rowL = threadIdx.x >> 2;  // 0..63
  int seg  = threadIdx.x & 3;   // 0..3
  uint32_t la = baseA + (uint32_t)(rowL * LDSTRIDE + seg * 8) * 4u;
  uint32_t lb = baseB + (uint32_t)(rowL * LDSTRIDE + seg * 8) * 4u;
  int mrowS = row0 + rowL;
  int iS = mrowS >> 1, bbS = mrowS & 1;
  const float* gbBase = Wo + (size_t)(col0 + rowL) * EMB + seg * 8;

  v8f acc0 = v8zero(), acc1 = v8zero();
  for (int k0 = 0; k0 < EMB; k0 += 32) {
    int e0 = k0 + seg * 8;
    int hS = e0 >> 6, dS = e0 & 63;
    const float* ga =
        outpre + (((size_t)(bbS * NH + hS)) * N_SEQ + iS) * HD + dS;
    __syncthreads();
    async_g2l_b128(la,      ga);
    async_g2l_b128(la + 16, ga + 4);
    async_g2l_b128(lb,      gbBase + k0);
    async_g2l_b128(lb + 16, gbBase + k0 + 4);
    wait_async0();
    __syncthreads();
#pragma unroll
    for (int kk = 0; kk < 32; kk += 4) {
      int dd = kk + 2 * half;
      v2f a  = *(const v2f*)&lA[(wr * 16 + r) * LDSTRIDE + dd];
      v2f b0 = *(const v2f*)&lB[(wc * 32 + r) * LDSTRIDE + dd];
      v2f b1 = *(const v2f*)&lB[(wc * 32 + 16 + r) * LDSTRIDE + dd];
      acc0 = wmma_f32(a, b0, acc0);
      acc1 = wmma_f32(a, b1, acc1);
    }
  }

#pragma unroll
  for (int ti = 0; ti < 2; ++ti) {
    int colg = col0 + wc * 32 + ti * 16 + r;
    float bval = bo[colg];
    v8f acc = ti ? acc1 : acc0;
#pragma unroll
    for (int t = 0; t < 8; ++t) {
      int mm = row0 + wr * 16 + t + 8 * half;
      out[(size_t)mm * EMB + colg] = acc[t] + bval;
    }
  }
}

// ---------------------------------------------------------------------------
// 11) aw_avg: mean over heads -> (b, n, P), appended after attn in d_out
// ---------------------------------------------------------------------------
__global__ void aw_avg_kernel(const float* __restrict__ awbuf,
                              float* __restrict__ dst) {
  int idx = blockIdx.x * blockDim.x + threadIdx.x;  // BATCH*N_SEQ*PQN = 65536
  int p  = idx & 15;
  int i  = (idx >> 4) & (N_SEQ - 1);
  int bb = idx >> 15;
  float acc = 0.f;
#pragma unroll
  for (int h = 0; h < NH; ++h)
    acc += awbuf[(((size_t)(bb * NH + h)) * N_SEQ + i) * PQN + p];
  dst[idx] = acc * (1.0f / NH);
}

// ---------------------------------------------------------------------------
extern "C" void kernel_launch(void* const* d_in, const int* in_sizes, int n_in,
                              void* d_out, int out_size, void* d_ws, size_t ws_size,
                              hipStream_t stream) {
  const float* query  = (const float*)d_in[0];
  const float* pquery = (const float*)d_in[1];
  const float* Wpq    = (const float*)d_in[2];
  const float* bpq    = (const float*)d_in[3];
  const float* Wk     = (const float*)d_in[4];
  const float* bk     = (const float*)d_in[5];
  const float* Wv     = (const float*)d_in[6];
  const float* bv     = (const float*)d_in[7];
  const float* Wo     = (const float*)d_in[8];
  const float* bo     = (const float*)d_in[9];

  float* out    = (float*)d_out;                      // attn (n,b,E) then aw_avg (b,n,P)
  float* aw_out = out + (size_t)N_SEQ * BATCH * EMB;

  float* ws     = (float*)d_ws;
  float* pq     = ws;                                 //    32768
  float* kbuf   = pq     + (size_t)BH * PQN * HD;     // 4194304
  float* vbuf   = kbuf   + (size_t)BH * N_SEQ * HD;   // 4194304
  float* eexp   = vbuf   + (size_t)BH * N_SEQ * HD;   // 1048576 (pattn -> Eexp in place)
  float* sexc   = eexp   + (size_t)BH * N_SEQ * PQN;  // 1048576
  float* awbuf  = sexc   + (size_t)BH * N_SEQ * PQN;  // 1048576
  float* gbuf   = awbuf  + (size_t)BH * N_SEQ * PQN;  // 4194304 (reused as mbuf)
  float* outpre = gbuf   + (size_t)BH * NC * HD * PQN;// 4194304
  float* mbuf   = gbuf;                               // reuse after stage 1

  // 1. pq projection
  pq_proj_kernel<<<128, 256, 0, stream>>>(pquery, Wpq, bpq, pq);
  // 2. K and V projections (64x16 blocks of 64x64 tiles, async-LDS staged)
  kv_proj_kernel<<<1024, 256, 0, stream>>>(query, Wk, bk, kbuf);
  kv_proj_kernel<<<1024, 256, 0, stream>>>(query, Wv, bv, vbuf);
  // 3. compressed logits pattn
  pattn_kernel<<<1024, 128, 0, stream>>>(query, pq, eexp);
  // 4. column max + exp + exclusive cumsum
  softmax_scan_kernel<<<BH * PQN, 256, 0, stream>>>(eexp, sexc);
  // 5-6. stage-1 chunk totals + exclusive prefix over chunks
  chunk_g_kernel<<<1024, 128, 0, stream>>>(kbuf, eexp, gbuf);
  chunk_scan_kernel<<<128, 256, 0, stream>>>(gbuf);
  // 7. stage-1 attn weights + softmax
  stage1_kernel<<<1024, 128, 0, stream>>>(query, kbuf, eexp, sexc, gbuf, awbuf);
  // 8-9. stage-2 chunk totals (reusing gbuf) + prefix + output accumulation
  chunk_m_kernel<<<1024, 128, 0, stream>>>(eexp, vbuf, mbuf);
  chunk_scan_kernel<<<128, 256, 0, stream>>>(mbuf);
  stage2_kernel<<<1024, 128, 0, stream>>>(awbuf, sexc, eexp, vbuf, mbuf, outpre);
  // 10. output projection
  out_proj_kernel<<<1024, 256, 0, stream>>>(outpre, Wo, bo, out);
  // 11. head-averaged attention weights
  aw_avg_kernel<<<256, 256, 0, stream>>>(awbuf, aw_out);
}